// PillarVFE_50783693308345
// MI455X (gfx1250) — compile-verified
//
#include <hip/hip_runtime.h>

// ---------------------------------------------------------------------------
// PillarVFE for MI455X (gfx1250, wave32).
// GEMM (32x10 @ 10x64 per pillar) runs on V_WMMA_F32_16X16X4_F32.
// ---------------------------------------------------------------------------

typedef float v2f __attribute__((ext_vector_type(2)));
typedef float v8f __attribute__((ext_vector_type(8)));

#define GY 1024
#define GX 1024
#define NUM_BEV 8
#define P 32
#define COUT 64

__device__ __forceinline__ float wave_sum(float v) {
#pragma unroll
  for (int o = 16; o > 0; o >>= 1) v += __shfl_xor(v, o, 32);
  return v;
}
__device__ __forceinline__ float wave_max(float v) {
#pragma unroll
  for (int o = 16; o > 0; o >>= 1) v = fmaxf(v, __shfl_xor(v, o, 32));
  return v;
}
__device__ __forceinline__ float wave_min(float v) {
#pragma unroll
  for (int o = 16; o > 0; o >>= 1) v = fminf(v, __shfl_xor(v, o, 32));
  return v;
}

// ---------------------------------------------------------------------------
// Kernel 1: zero the vox_bev region (float4 stores, grid-stride).
// ---------------------------------------------------------------------------
__global__ __launch_bounds__(256) void bev_zero_kernel(float4* __restrict__ p,
                                                       long n4) {
  long i = (long)blockIdx.x * blockDim.x + threadIdx.x;
  long stride = (long)gridDim.x * blockDim.x;
  float4 z = make_float4(0.f, 0.f, 0.f, 0.f);
  for (; i < n4; i += stride) p[i] = z;
}

// ---------------------------------------------------------------------------
// Kernel 2: one wave32 per pillar.
// ---------------------------------------------------------------------------
__global__ __launch_bounds__(256) void pillar_vfe_kernel(
    const float* __restrict__ vf,        // (N, 32, 4)
    const float* __restrict__ W,         // (10, 64)
    const float* __restrict__ gamma,     // (64)
    const float* __restrict__ beta,      // (64)
    const float* __restrict__ rmean,     // (64)
    const float* __restrict__ rvar,      // (64)
    const int* __restrict__ vnp,         // (N)
    const int* __restrict__ coords,      // (N, 4)
    float* __restrict__ out_pf,          // (N, 64)
    float* __restrict__ out_bev,         // (bs, 8, GY, GX)
    int N, int batch_size) {
  // voxel grid constants
  const float VXc = 0.1f, VYc = 0.1f, VZc = 4.0f;
  const float X_OFF = VXc * 0.5f - 51.2f;
  const float Y_OFF = VYc * 0.5f - 51.2f;
  const float Z_OFF = VZc * 0.5f - 3.0f;

  __shared__ float sW[12][COUT];        // K padded 10 -> 12
  __shared__ float sScale[COUT];
  __shared__ float sBias[COUT];
  __shared__ float sF[8][P][13];        // per-wave feature tile, stride 13

  const int lane = threadIdx.x & 31;
  const int wv   = threadIdx.x >> 5;

  // --- block-wide setup: W and folded BN params into LDS --------------------
  for (int t = threadIdx.x; t < 12 * COUT; t += 256) {
    int r = t >> 6, c = t & 63;
    sW[r][c] = (r < 10) ? W[r * COUT + c] : 0.f;
  }
  if (threadIdx.x < COUT) {
    int c = threadIdx.x;
    float s = gamma[c] * rsqrtf(rvar[c] + 1e-3f);
    sScale[c] = s;
    sBias[c]  = beta[c] - rmean[c] * s;
  }
  __syncthreads();

  // --- hoist B fragments (shared by every pillar this wave handles) ---------
  // B layout (4x16 KxN, mirrors 16x4 A): VGPR0 = K {0|2}, VGPR1 = K {1|3};
  // lanes 0-15 -> K+0/1, lanes 16-31 -> K+2/3; N = lane & 15.
  const int khalf = (lane >> 4) << 1;   // 0 for lanes 0-15, 2 for 16-31
  v2f bf[4][3];
#pragma unroll
  for (int nt = 0; nt < 4; ++nt) {
#pragma unroll
    for (int k = 0; k < 3; ++k) {
      int kk  = k * 4 + khalf;
      int col = nt * 16 + (lane & 15);
      v2f b;
      b.x = sW[kk][col];
      b.y = sW[kk + 1][col];
      bf[nt][k] = b;
    }
  }

  // zero the K-padding columns of this wave's feature tile (written once)
  sF[wv][lane][10] = 0.f;
  sF[wv][lane][11] = 0.f;

  const int waveId   = blockIdx.x * 8 + wv;
  const int waveStep = gridDim.x * 8;

  for (int n = waveId; n < N; n += waveStep) {
    // ---- per-point load (coalesced: 512B contiguous per pillar) -----------
    const float4 p = ((const float4*)vf)[(long)n * P + lane];
    const int   np = vnp[n];
    const float fn = (float)np;
    const float safe_n = fmaxf(fn, 1.0f);
    const float m = (lane < np) ? 1.f : 0.f;

    const int4 cd = ((const int4*)coords)[n];   // (b, zbin, y, x)
    const float cx = (float)cd.w * VXc + X_OFF;
    const float cy = (float)cd.z * VYc + Y_OFF;
    const float cz = (float)cd.y * VZc + Z_OFF;

    // ---- unmasked mean over ALL P points (reference semantics) ------------
    const float mean_x = wave_sum(p.x) / fn;
    const float mean_y = wave_sum(p.y) / fn;
    const float mean_z = wave_sum(p.z) / fn;

    // ---- build masked 10-channel features into LDS ------------------------
    float* fr = &sF[wv][lane][0];
    fr[0] = p.x * m;
    fr[1] = p.y * m;
    fr[2] = p.z * m;
    fr[3] = p.w * m;
    fr[4] = (p.x - mean_x) * m;
    fr[5] = (p.y - mean_y) * m;
    fr[6] = (p.z - mean_z) * m;
    fr[7] = (p.x - cx) * m;
    fr[8] = (p.y - cy) * m;
    fr[9] = (p.z - cz) * m;
    __builtin_amdgcn_wave_barrier();   // keep LDS write->read ordered in-wave

    // ---- GEMM: (32x12) @ (12x64) via V_WMMA_F32_16X16X4_F32 ---------------
    v8f acc[2][4];
#pragma unroll
    for (int mt = 0; mt < 2; ++mt)
#pragma unroll
      for (int nt = 0; nt < 4; ++nt) acc[mt][nt] = (v8f)(0.f);

#pragma unroll
    for (int mt = 0; mt < 2; ++mt) {
      const int row = mt * 16 + (lane & 15);
#pragma unroll
      for (int k = 0; k < 3; ++k) {
        const int kk = k * 4 + khalf;
        v2f a;
        a.x = sF[wv][row][kk];
        a.y = sF[wv][row][kk + 1];
#pragma unroll
        for (int nt = 0; nt < 4; ++nt) {
          acc[mt][nt] = __builtin_amdgcn_wmma_f32_16x16x4_f32(
              false, a, false, bf[nt][k], (short)0, acc[mt][nt], false, false);
        }
      }
    }

    // ---- BN + ReLU + max over the 32 points -------------------------------
    // C layout: VGPR v holds row (mt*16 + v + 8*(lane>=16)), col = nt*16+(lane&15)
#pragma unroll
    for (int nt = 0; nt < 4; ++nt) {
      const int col = nt * 16 + (lane & 15);
      const float sc = sScale[col];
      const float bi = sBias[col];
      float mx = 0.f;   // ReLU outputs are >= 0, so 0 is the identity here
#pragma unroll
      for (int v = 0; v < 8; ++v) {
        mx = fmaxf(mx, fmaxf(acc[0][nt][v] * sc + bi, 0.f));
        mx = fmaxf(mx, fmaxf(acc[1][nt][v] * sc + bi, 0.f));
      }
      mx = fmaxf(mx, __shfl_xor(mx, 16, 32));  // lanes l and l+16: same column
      if (lane < 16) out_pf[(long)n * COUT + col] = mx;
    }

    // ---- 8 BEV statistics (masked reductions) -----------------------------
    const float sxm = wave_sum(p.x * m);
    const float sym = wave_sum(p.y * m);
    const float szm = wave_sum(p.z * m);
    const float sim = wave_sum(p.w * m);
    const float zmax = wave_max(p.z * m + (1.f - m) * -1.0e6f);
    const float zmin = wave_min(p.z * m + (1.f - m) * 1.0e6f);

    const float pmx = sxm / safe_n, pmy = sym / safe_n, pmz = szm / safe_n;
    const float dx = p.x - pmx, dy = p.y - pmy, dz = p.z - pmz;
    const float var_x = wave_sum(dx * dx * m) / safe_n;
    const float var_y = wave_sum(dy * dy * m) / safe_n;
    const float var_z = wave_sum(dz * dz * m) / safe_n;

    float bev = fn * (1.0f / (float)P);              // num_points_norm
    bev = (lane == 1) ? (sim / safe_n)   : bev;      // mean_intensity
    bev = (lane == 2) ? (szm / safe_n)   : bev;      // mean_height
    bev = (lane == 3) ? zmax             : bev;      // max_height
    bev = (lane == 4) ? (zmax - zmin)    : bev;      // height_span
    bev = (lane == 5) ? var_x            : bev;
    bev = (lane == 6) ? var_y            : bev;
    bev = (lane == 7) ? var_z            : bev;

    int b = cd.x;
    b = (b < 0) ? 0 : b;
    b = (b > batch_size - 1) ? (batch_size - 1) : b;
    if (lane < NUM_BEV) {
      long idx = (((long)b * NUM_BEV + lane) * GY + cd.z) * GX + cd.w;
      out_bev[idx] = bev;
    }
  }
}

extern "C" void kernel_launch(void* const* d_in, const int* in_sizes, int n_in,
                              void* d_out, int out_size, void* d_ws,
                              size_t ws_size, hipStream_t stream) {
  const float* vf     = (const float*)d_in[0];
  const float* W      = (const float*)d_in[1];
  const float* gamma  = (const float*)d_in[2];
  const float* beta   = (const float*)d_in[3];
  const float* rmean  = (const float*)d_in[4];
  const float* rvar   = (const float*)d_in[5];
  const int*   vnp    = (const int*)d_in[6];
  const int*   coords = (const int*)d_in[7];

  const int N  = in_sizes[6];       // number of pillars
  const int bs = in_sizes[8];       // record_len length == batch size

  float* out_pf  = (float*)d_out;
  float* out_bev = out_pf + (long)N * COUT;
  const long bev_elems = (long)out_size - (long)N * COUT;
  const long n4 = bev_elems >> 2;   // region is 16B-aligned (N*64*4 % 16 == 0)

  // 1) zero vox_bev
  bev_zero_kernel<<<2048, 256, 0, stream>>>((float4*)out_bev, n4);

  // 2) pillar kernel: 8 wave32 per block, one pillar per wave (grid-stride)
  int waves_needed = (N + 0) ;                   // one pillar per wave
  int blocks = (waves_needed + 7) / 8;
  if (blocks > 2500) blocks = 2500;              // grid-stride covers the rest
  pillar_vfe_kernel<<<blocks, 256, 0, stream>>>(vf, W, gamma, beta, rmean,
                                                rvar, vnp, coords, out_pf,
                                                out_bev, N, bs);
}